// RegistrationLoss_81638738362420
// MI455X (gfx1250) — compile-verified
//
#include <hip/hip_runtime.h>
#include <hip/hip_bf16.h>

// ---------------------------------------------------------------------------
// RegistrationLoss for MI455X (gfx1250, wave32).
//
// Strategy (bandwidth-bound, ~0.85 GB total traffic -> ~37us @ 23.3 TB/s):
//   Pass 1: W-axis 9-wide box filter of {I, J, I*J, I*I, J*J} expressed as a
//           banded GEMM and executed on the matrix core with
//           V_WMMA_F32_16X16X4_F32 (f32 WMMA => bit-identical to scalar adds).
//   Pass 2: H-axis box filter, in-place, register sliding window, coalesced.
//   Pass 3: D-axis box filter fused with NCC formula + block reduction.
//   Pass 4: bending-energy grid-stride reduction over flow.
//   Pass 5: deterministic single-block finalize -> (total, L_sim, L_reg).
//
// Workspace layout (floats): 5 field volumes of VOL each, then partials:
//   [0..199]   ncc per-block partials
//   [256..]    bending per-block partials (512 blocks x 3 sums)
// Requires ws_size >= 5*VOL*4 + 4096*4 bytes (~163.9 MB).
// ---------------------------------------------------------------------------

typedef float v2f __attribute__((ext_vector_type(2)));
typedef float v8f __attribute__((ext_vector_type(8)));

constexpr int Bn = 2, Dn = 160, Hn = 160, Wn = 160;
constexpr int VOL   = Bn * Dn * Hn * Wn;   // 8,192,000 per field
constexpr int NROWS = Bn * Dn * Hn;        // 51,200 W-lines
constexpr int NCC_N = VOL;                 // ncc mean denominator
constexpr float BEND_CNT = 24422400.0f;    // B*3*(D-1)*H*W (== gy,gx counts)

constexpr int HCOLS = 5 * Bn * Dn * Wn;    // 256,000 columns in pass 2
constexpr int DCOLS = Bn * Hn * Wn;        // 51,200 columns in pass 3
constexpr int BEND_TOT = Bn * 3 * Dn * Hn * Wn; // 24,576,000
constexpr int BEND_BLOCKS = 512;
constexpr int NCC_BLOCKS  = DCOLS / 256;   // 200

// --------------------------- Pass 1: W filter (WMMA) -----------------------
// out[m, c+n] = sum_{j=c+n-4}^{c+n+4} in[m, j]   (zero padded)
// As GEMM: Out16x16 = In[16 x 24] * Band[24 x 16],
//   Band[k, n] = 1 iff 0 <= k - n <= 8   (k = chunk-global K index, n = col)
// Six K=4 chunks per tile via V_WMMA_F32_16X16X4_F32.
//
// Operand layouts (ISA 7.12.2, wave32):
//   A (16x4 f32):  lane L holds A[L%16, 2*(L/16)+{0,1}] in {v0,v1}
//   B (4x16 f32):  assumed to mirror A's K pairing:
//                  lane L holds B[2*(L/16)+{0,1}, L%16] in {v0,v1}
//   C/D (16x16):   vgpr v, lane L -> D[v + 8*(L/16), L%16]

__device__ __forceinline__ float ldz(const float* s, int m, int col) {
    return ((unsigned)col < (unsigned)Wn) ? s[m * Wn + col] : 0.0f;
}

__global__ __launch_bounds__(32)
void wfilt_wmma_kernel(const float* __restrict__ I, const float* __restrict__ J,
                       float* __restrict__ F) {
    __shared__ float sI[16 * Wn];
    __shared__ float sJ[16 * Wn];

    const int lane    = threadIdx.x;          // 0..31
    const int rowBase = blockIdx.x * 16;      // 16 independent W-lines
    const size_t gbase = (size_t)rowBase * Wn;

    for (int idx = lane; idx < 16 * Wn; idx += 32) {
        sI[idx] = I[gbase + idx];
        sJ[idx] = J[gbase + idx];
    }
    __syncthreads();

    const int m    = lane & 15;   // A row / B col / D col
    const int half = lane >> 4;
    const int koff = half * 2;    // K pair selected by lane half

    float* __restrict__ f0 = F;
    float* __restrict__ f1 = F + (size_t)VOL;
    float* __restrict__ f2 = F + (size_t)2 * VOL;
    float* __restrict__ f3 = F + (size_t)3 * VOL;
    float* __restrict__ f4 = F + (size_t)4 * VOL;

    for (int tile = 0; tile < Wn / 16; ++tile) {
        const int c0 = tile * 16;
        v8f a0 = {}; v8f a1 = {}; v8f a2 = {}; v8f a3 = {}; v8f a4 = {};

#pragma unroll
        for (int q = 0; q < 6; ++q) {
            const int j0 = c0 - 4 + q * 4 + koff;  // global input column
            const float iv0 = ldz(sI, m, j0);
            const float iv1 = ldz(sI, m, j0 + 1);
            const float jv0 = ldz(sJ, m, j0);
            const float jv1 = ldz(sJ, m, j0 + 1);

            // Band operand: k_chunk = 4q + {koff, koff+1}, n = m
            const int k0 = q * 4 + koff - m;
            v2f bv = { (k0 >= 0 && k0 <= 8) ? 1.0f : 0.0f,
                       (k0 + 1 >= 0 && k0 + 1 <= 8) ? 1.0f : 0.0f };

            v2f aI  = { iv0,       iv1 };
            v2f aJ  = { jv0,       jv1 };
            v2f aIJ = { iv0 * jv0, iv1 * jv1 };
            v2f aII = { iv0 * iv0, iv1 * iv1 };
            v2f aJJ = { jv0 * jv0, jv1 * jv1 };

            a0 = __builtin_amdgcn_wmma_f32_16x16x4_f32(false, aI,  false, bv, (short)0, a0, false, false);
            a1 = __builtin_amdgcn_wmma_f32_16x16x4_f32(false, aJ,  false, bv, (short)0, a1, false, false);
            a2 = __builtin_amdgcn_wmma_f32_16x16x4_f32(false, aIJ, false, bv, (short)0, a2, false, false);
            a3 = __builtin_amdgcn_wmma_f32_16x16x4_f32(false, aII, false, bv, (short)0, a3, false, false);
            a4 = __builtin_amdgcn_wmma_f32_16x16x4_f32(false, aJJ, false, bv, (short)0, a4, false, false);
        }

#pragma unroll
        for (int v = 0; v < 8; ++v) {
            const size_t o = (size_t)(rowBase + v + 8 * half) * Wn + c0 + m;
            f0[o] = a0[v];
            f1[o] = a1[v];
            f2[o] = a2[v];
            f3[o] = a3[v];
            f4[o] = a4[v];
        }
    }
}

// --------------------------- Pass 2: H filter (in place) -------------------
#define WIN_DECL(p) float p##0=0,p##1=0,p##2=0,p##3=0,p##4=0,p##5=0,p##6=0,p##7=0,p##8=0, p##s=0
#define WIN_STEP(p, xv) do { p##s += (xv) - p##0; \
    p##0=p##1; p##1=p##2; p##2=p##3; p##3=p##4; p##4=p##5; \
    p##5=p##6; p##6=p##7; p##7=p##8; p##8=(xv); } while (0)

__global__ __launch_bounds__(256)
void hfilt_kernel(float* __restrict__ F) {
    const int t = blockIdx.x * blockDim.x + threadIdx.x;
    if (t >= HCOLS) return;
    const int w = t % Wn;
    const int d = (t / Wn) % Dn;
    const int b = (t / (Wn * Dn)) % Bn;
    const int f = t / (Wn * Dn * Bn);
    float* base = F + (size_t)f * VOL + ((size_t)(b * Dn + d) * Hn) * Wn + w;

    WIN_DECL(u);
    for (int i = 0; i < Hn + 4; ++i) {
        const float x = (i < Hn) ? base[(size_t)i * Wn] : 0.0f;
        WIN_STEP(u, x);
        if (i >= 4) base[(size_t)(i - 4) * Wn] = us;
    }
}

// ------------------- Pass 3: D filter fused with NCC + reduce --------------
__device__ __forceinline__ float block_reduce_sum(float v, float* sm) {
    const int t = threadIdx.x;
    sm[t] = v;
    __syncthreads();
    for (int o = 128; o > 0; o >>= 1) {
        if (t < o) sm[t] += sm[t + o];
        __syncthreads();
    }
    const float r = sm[0];
    __syncthreads();
    return r;
}

__global__ __launch_bounds__(256)
void dfilt_ncc_kernel(const float* __restrict__ F, float* __restrict__ nccPart) {
    __shared__ float sm[256];
    const int t = blockIdx.x * blockDim.x + threadIdx.x;   // < DCOLS
    const int w = t % Wn;
    const int h = (t / Wn) % Hn;
    const int b = t / (Wn * Hn);
    const size_t col = ((size_t)(b * Dn) * Hn + h) * Wn + w;  // d = 0
    const size_t step = (size_t)Hn * Wn;

    const float* p0 = F + col;
    const float* p1 = F + (size_t)VOL + col;
    const float* p2 = F + (size_t)2 * VOL + col;
    const float* p3 = F + (size_t)3 * VOL + col;
    const float* p4 = F + (size_t)4 * VOL + col;

    WIN_DECL(a); WIN_DECL(c); WIN_DECL(e); WIN_DECL(g); WIN_DECL(k);
    float ncc_acc = 0.0f;

    for (int i = 0; i < Dn + 4; ++i) {
        const bool in = (i < Dn);
        const size_t o = (size_t)(in ? i : 0) * step;
        const float x0 = in ? p0[o] : 0.0f;
        const float x1 = in ? p1[o] : 0.0f;
        const float x2 = in ? p2[o] : 0.0f;
        const float x3 = in ? p3[o] : 0.0f;
        const float x4 = in ? p4[o] : 0.0f;
        WIN_STEP(a, x0); WIN_STEP(c, x1); WIN_STEP(e, x2);
        WIN_STEP(g, x3); WIN_STEP(k, x4);
        if (i >= 4) {
            const float muI = as,  muJ = cs;
            const float cross = es - muI * muJ;
            const float Ivar  = gs - muI * muI;
            const float Jvar  = ks - muJ * muJ;
            ncc_acc += cross / (sqrtf(Ivar * Jvar) + 1e-5f);
        }
    }

    const float tot = block_reduce_sum(ncc_acc, sm);
    if (threadIdx.x == 0) nccPart[blockIdx.x] = tot;
}

// --------------------------- Pass 4: bending energy ------------------------
__global__ __launch_bounds__(256)
void bend_kernel(const float* __restrict__ flow, float* __restrict__ bendPart) {
    __shared__ float sm[256];
    float sz = 0.0f, sy = 0.0f, sx = 0.0f;
    const int stride = gridDim.x * blockDim.x;
    for (int e = blockIdx.x * blockDim.x + threadIdx.x; e < BEND_TOT; e += stride) {
        const int w = e % Wn;
        const int h = (e / Wn) % Hn;
        const int d = (e / (Wn * Hn)) % Dn;
        const float v = flow[e];
        if (d + 1 < Dn) { const float g = flow[e + Hn * Wn] - v; sz += g * g; }
        if (h + 1 < Hn) { const float g = flow[e + Wn] - v;      sy += g * g; }
        if (w + 1 < Wn) { const float g = flow[e + 1] - v;       sx += g * g; }
    }
    sz = block_reduce_sum(sz, sm);
    sy = block_reduce_sum(sy, sm);
    sx = block_reduce_sum(sx, sm);
    if (threadIdx.x == 0) {
        bendPart[blockIdx.x * 3 + 0] = sz;
        bendPart[blockIdx.x * 3 + 1] = sy;
        bendPart[blockIdx.x * 3 + 2] = sx;
    }
}

// --------------------------- Pass 5: finalize ------------------------------
__global__ void finalize_kernel(const float* __restrict__ part,
                                float* __restrict__ out) {
    if (threadIdx.x != 0 || blockIdx.x != 0) return;
    float sn = 0.0f;
    for (int i = 0; i < NCC_BLOCKS; ++i) sn += part[i];
    float sz = 0.0f, sy = 0.0f, sx = 0.0f;
    const float* bp = part + 256;
    for (int b = 0; b < BEND_BLOCKS; ++b) {
        sz += bp[b * 3 + 0];
        sy += bp[b * 3 + 1];
        sx += bp[b * 3 + 2];
    }
    const float l_sim = -(sn / (float)NCC_N);
    const float l_reg = (sz + sy + sx) / BEND_CNT; // equal counts per axis
    out[0] = l_sim + l_reg;
    out[1] = l_sim;
    out[2] = l_reg;
}

// --------------------------- launcher --------------------------------------
extern "C" void kernel_launch(void* const* d_in, const int* in_sizes, int n_in,
                              void* d_out, int out_size, void* d_ws, size_t ws_size,
                              hipStream_t stream) {
    const float* I    = (const float*)d_in[0];  // fixed_image  [2,1,160,160,160]
    const float* J    = (const float*)d_in[1];  // warped_image [2,1,160,160,160]
    const float* flow = (const float*)d_in[2];  // flow         [2,3,160,160,160]
    float* out = (float*)d_out;                 // (total, L_sim, L_reg)

    float* F    = (float*)d_ws;                 // 5 field volumes
    float* part = F + (size_t)5 * VOL;          // [0..199] ncc, [256..] bending

    wfilt_wmma_kernel<<<NROWS / 16, 32, 0, stream>>>(I, J, F);
    hfilt_kernel<<<HCOLS / 256, 256, 0, stream>>>(F);
    dfilt_ncc_kernel<<<NCC_BLOCKS, 256, 0, stream>>>(F, part);
    bend_kernel<<<BEND_BLOCKS, 256, 0, stream>>>(flow, part + 256);
    finalize_kernel<<<1, 32, 0, stream>>>(part, out);
}